// MFConv_936302871077
// MI455X (gfx1250) — compile-verified
//
#include <hip/hip_runtime.h>

typedef __attribute__((ext_vector_type(2))) float v2f;
typedef __attribute__((ext_vector_type(8))) float v8f;

#define IN_F   128
#define OUT_F  128
#define MAXDEG 4
#define NBUCK  5
#define TILE_M 16
#define LDA    260   // padded row pitch in floats (kills 16-way LDS bank conflicts)
#define SCAN_B 256

// ---------------- init helpers ----------------
__global__ void fill_i32_kernel(int* __restrict__ p, int val, size_t n) {
    size_t i = (size_t)blockIdx.x * blockDim.x + threadIdx.x;
    size_t stride = (size_t)gridDim.x * blockDim.x;
    for (; i < n; i += stride) p[i] = val;
}

// ---------------- in-degree (int) ----------------
__global__ void degree_kernel(const int* __restrict__ col, int* __restrict__ degi, int E) {
    int e = blockIdx.x * blockDim.x + threadIdx.x;
    if (e < E) atomicAdd(&degi[col[e]], 1);
}

// ---------------- two-level exclusive scan over degi -> rowstart ----------------
__global__ void scan1_kernel(const int* __restrict__ degi, int* __restrict__ rowstart,
                             int* __restrict__ bsum, int N) {
    __shared__ int sh[SCAN_B];
    int i = blockIdx.x * SCAN_B + threadIdx.x;
    int v = (i < N) ? degi[i] : 0;
    sh[threadIdx.x] = v;
    __syncthreads();
    #pragma unroll
    for (int off = 1; off < SCAN_B; off <<= 1) {
        int t = (threadIdx.x >= off) ? sh[threadIdx.x - off] : 0;
        __syncthreads();
        sh[threadIdx.x] += t;
        __syncthreads();
    }
    if (i < N) rowstart[i] = sh[threadIdx.x] - v;          // exclusive within block
    if (threadIdx.x == SCAN_B - 1) bsum[blockIdx.x] = sh[SCAN_B - 1];
}

__global__ void scan2_kernel(int* __restrict__ bsum, int nBlocks) {
    if (threadIdx.x == 0 && blockIdx.x == 0) {
        int acc = 0;
        for (int b = 0; b < nBlocks; ++b) { int t = bsum[b]; bsum[b] = acc; acc += t; }
    }
}

__global__ void scan3_kernel(int* __restrict__ rowstart, const int* __restrict__ bsum, int N) {
    int i = blockIdx.x * SCAN_B + threadIdx.x;
    if (i < N) rowstart[i] += bsum[blockIdx.x];
}

// ---------------- CSR fill: edge_src[rowstart[dst] + slot] = src ----------------
__global__ void csr_fill_kernel(const int* __restrict__ row, const int* __restrict__ col,
                                const int* __restrict__ rowstart, int* __restrict__ cursor,
                                int* __restrict__ edge_src, int E) {
    int e = blockIdx.x * blockDim.x + threadIdx.x;
    if (e >= E) return;
    int d = col[e];
    int pos = atomicAdd(&cursor[d], 1);
    edge_src[rowstart[d] + pos] = row[e];
}

// ---------------- bucket counting sort (meta: [0..4]=counts [5..9]=offsets [10..14]=cursors) --
__global__ void bucket_count_kernel(const int* __restrict__ degi, int* __restrict__ meta, int N) {
    int i = blockIdx.x * blockDim.x + threadIdx.x;
    if (i >= N) return;
    int b = min(degi[i], MAXDEG);
    atomicAdd(&meta[b], 1);
}

__global__ void bucket_offset_kernel(int* __restrict__ meta) {
    if (threadIdx.x == 0 && blockIdx.x == 0) {
        int off = 0;
        for (int b = 0; b < NBUCK; ++b) {
            meta[5 + b] = off;
            off += (meta[b] + TILE_M - 1) & ~(TILE_M - 1);  // 16-aligned segments
        }
    }
}

__global__ void bucket_scatter_kernel(const int* __restrict__ degi, int* __restrict__ meta,
                                      int* __restrict__ perm, int N) {
    int i = blockIdx.x * blockDim.x + threadIdx.x;
    if (i >= N) return;
    int b = min(degi[i], MAXDEG);
    int pos = atomicAdd(&meta[10 + b], 1);
    perm[meta[5 + b] + pos] = i;
}

// ---------------- fused gather-mean + per-bucket dual-GEMM (V_WMMA_F32_16X16X4_F32) --------
// Block = one 16-node tile (homogeneous bucket). 8 waves x 16 out-cols = 128 outputs.
__global__ __launch_bounds__(256) void mfconv_gemm_kernel(
    const float* __restrict__ x,
    const int* __restrict__ degi, const int* __restrict__ rowstart,
    const int* __restrict__ edge_src, const int* __restrict__ perm,
    const float* __restrict__ neigh_w, const float* __restrict__ neigh_b,
    const float* __restrict__ root_w,  const float* __restrict__ root_b,
    float* __restrict__ out)
{
    __shared__ float ldsA[TILE_M * LDA];
    __shared__ int   s_node[TILE_M];
    __shared__ int   s_bucket;

    const int tid  = threadIdx.x;
    const int tile = blockIdx.x;

    if (tid < TILE_M) s_node[tid] = perm[tile * TILE_M + tid];
    __syncthreads();

    if (tid == 0) {
        int b = -1;
        for (int i = 0; i < TILE_M; ++i) {
            int nd = s_node[i];
            if (nd >= 0) { b = min(degi[nd], MAXDEG); break; }
        }
        s_bucket = b;
    }

    // Stage A = [mean(neigh) | x] (16 x 256) into LDS; neighbor gather fused here.
    // Thread t handles node m = t/16, 16-feature chunk (t%16)*16.
    {
        const int m    = tid >> 4;
        const int k0   = (tid & 15) * 16;
        const int node = s_node[m];
        float4 a0 = make_float4(0.f,0.f,0.f,0.f), a1 = a0, a2 = a0, a3 = a0;
        if (node >= 0) {
            if (k0 < IN_F) {                       // neigh-mean half: gather neighbors
                const int start = rowstart[node];
                const int d     = degi[node];
                for (int j = 0; j < d; ++j) {
                    const float* xs = x + (size_t)edge_src[start + j] * IN_F + k0;
                    const float4 v0 = *(const float4*)(xs + 0);
                    const float4 v1 = *(const float4*)(xs + 4);
                    const float4 v2 = *(const float4*)(xs + 8);
                    const float4 v3 = *(const float4*)(xs + 12);
                    a0.x += v0.x; a0.y += v0.y; a0.z += v0.z; a0.w += v0.w;
                    a1.x += v1.x; a1.y += v1.y; a1.z += v1.z; a1.w += v1.w;
                    a2.x += v2.x; a2.y += v2.y; a2.z += v2.z; a2.w += v2.w;
                    a3.x += v3.x; a3.y += v3.y; a3.z += v3.z; a3.w += v3.w;
                }
                const float inv = 1.0f / fmaxf((float)d, 1.0f);
                a0.x *= inv; a0.y *= inv; a0.z *= inv; a0.w *= inv;
                a1.x *= inv; a1.y *= inv; a1.z *= inv; a1.w *= inv;
                a2.x *= inv; a2.y *= inv; a2.z *= inv; a2.w *= inv;
                a3.x *= inv; a3.y *= inv; a3.z *= inv; a3.w *= inv;
            } else {                               // root half: copy x[node]
                const float* xs = x + (size_t)node * IN_F + (k0 - IN_F);
                a0 = *(const float4*)(xs + 0);
                a1 = *(const float4*)(xs + 4);
                a2 = *(const float4*)(xs + 8);
                a3 = *(const float4*)(xs + 12);
            }
        }
        float* dst = &ldsA[m * LDA + k0];
        *(float4*)(dst + 0)  = a0;
        *(float4*)(dst + 4)  = a1;
        *(float4*)(dst + 8)  = a2;
        *(float4*)(dst + 12) = a3;
    }
    __syncthreads();

    const int bucket = s_bucket;
    if (bucket < 0) return;  // uniform: whole tile is padding -> EXEC stays full below

    const int wave  = tid >> 5;
    const int lane  = tid & 31;
    const int lr    = lane & 15;
    const int khalf = (lane >> 4) << 1;      // lanes 0-15 -> K,K+1 ; lanes 16-31 -> K+2,K+3
    const int ncol  = wave * 16 + lr;        // output column

    const float* __restrict__ wn   = neigh_w + ((size_t)bucket * OUT_F + ncol) * IN_F;
    const float* __restrict__ wr   = root_w  + ((size_t)bucket * OUT_F + ncol) * IN_F;
    const float* __restrict__ aRow = &ldsA[lr * LDA];

    v8f acc = {};
    #pragma unroll
    for (int kk = 0; kk < 32; ++kk) {        // neigh half of cat-K
        const int k = kk * 4 + khalf;
        v2f a, b;
        a.x = aRow[k];     a.y = aRow[k + 1];
        b.x = wn[k];       b.y = wn[k + 1];
        acc = __builtin_amdgcn_wmma_f32_16x16x4_f32(false, a, false, b,
                                                    (short)0, acc, false, false);
    }
    #pragma unroll
    for (int kk = 0; kk < 32; ++kk) {        // root half of cat-K
        const int k = kk * 4 + khalf;
        v2f a, b;
        a.x = aRow[IN_F + k]; a.y = aRow[IN_F + k + 1];
        b.x = wr[k];          b.y = wr[k + 1];
        acc = __builtin_amdgcn_wmma_f32_16x16x4_f32(false, a, false, b,
                                                    (short)0, acc, false, false);
    }

    const float bias = neigh_b[bucket * OUT_F + ncol] + root_b[bucket * OUT_F + ncol];
    const int mbase  = (lane >> 4) * 8;      // C/D layout: VGPR i <-> M=i (lanes 0-15) / M=8+i
    #pragma unroll
    for (int i = 0; i < 8; ++i) {
        const int node = s_node[mbase + i];
        if (node >= 0) out[(size_t)node * OUT_F + ncol] = acc[i] + bias;
    }
}

// ---------------- launch ----------------
extern "C" void kernel_launch(void* const* d_in, const int* in_sizes, int n_in,
                              void* d_out, int out_size, void* d_ws, size_t ws_size,
                              hipStream_t stream) {
    const float* x       = (const float*)d_in[0];
    const float* neigh_w = (const float*)d_in[1];
    const float* neigh_b = (const float*)d_in[2];
    const float* root_w  = (const float*)d_in[3];
    const float* root_b  = (const float*)d_in[4];
    const int*   eidx    = (const int*)d_in[5];

    const int N = in_sizes[0] / IN_F;
    const int E = in_sizes[5] / 2;
    const int* row = eidx;
    const int* col = eidx + E;

    const int nScanBlocks = (N + SCAN_B - 1) / SCAN_B;
    const int permLen = ((N + NBUCK * TILE_M + TILE_M - 1) / TILE_M) * TILE_M;
    const int nTiles  = permLen / TILE_M;

    // workspace carve-up (all int32): degi | cursor | rowstart | bsum | edge_src | perm | meta
    char* ws = (char*)d_ws;
    int* degi     = (int*)ws;  ws += (size_t)N * sizeof(int);
    int* cursor   = (int*)ws;  ws += (size_t)N * sizeof(int);
    int* rowstart = (int*)ws;  ws += (size_t)N * sizeof(int);
    int* bsum     = (int*)ws;  ws += (size_t)((nScanBlocks + 15) & ~15) * sizeof(int);
    int* edge_src = (int*)ws;  ws += (size_t)E * sizeof(int);
    int* perm     = (int*)ws;  ws += (size_t)permLen * sizeof(int);
    int* meta     = (int*)ws;  // 16 ints

    // init: degi & cursor contiguous -> one zero pass; perm = -1; meta = 0
    fill_i32_kernel<<<1024, 256, 0, stream>>>(degi, 0, (size_t)2 * N);
    fill_i32_kernel<<<1024, 256, 0, stream>>>(perm, -1, (size_t)permLen);
    fill_i32_kernel<<<1, 32, 0, stream>>>(meta, 0, 16);

    // CSR build: degree -> exclusive scan -> cursor fill
    degree_kernel<<<(E + 255) / 256, 256, 0, stream>>>(col, degi, E);
    scan1_kernel<<<nScanBlocks, SCAN_B, 0, stream>>>(degi, rowstart, bsum, N);
    scan2_kernel<<<1, 1, 0, stream>>>(bsum, nScanBlocks);
    scan3_kernel<<<nScanBlocks, SCAN_B, 0, stream>>>(rowstart, bsum, N);
    csr_fill_kernel<<<(E + 255) / 256, 256, 0, stream>>>(row, col, rowstart, cursor,
                                                         edge_src, E);

    // bucket counting sort into 16-aligned segments
    bucket_count_kernel<<<(N + 255) / 256, 256, 0, stream>>>(degi, meta, N);
    bucket_offset_kernel<<<1, 1, 0, stream>>>(meta);
    bucket_scatter_kernel<<<(N + 255) / 256, 256, 0, stream>>>(degi, meta, perm, N);

    // fused gather-mean + dual GEMM + bias + scatter-out
    mfconv_gemm_kernel<<<nTiles, 256, 0, stream>>>(x, degi, rowstart, edge_src, perm,
                                                   neigh_w, neigh_b, root_w, root_b,
                                                   (float*)d_out);
}